// ScaledDotProductAttention1_33689723470285
// MI455X (gfx1250) — compile-verified
//
#include <hip/hip_runtime.h>
#include <hip/hip_bf16.h>

typedef __attribute__((ext_vector_type(2))) float v2f;
typedef __attribute__((ext_vector_type(8))) float v8f;

#define SEQ 20
#define DK 64
#define WAVES_PER_WG 4
#define QK_STRIDE 68   // floats; fragment banks = (4*row + d) % 64 -> conflict-free
#define V_STRIDE  72   // floats; 2*stride % 64 = 16 -> half-wave bank sets disjoint
#define P_STRIDE  36   // floats; 32x32 zero-padded (attn + Md) matrix
// All matrix slices padded to 32 rows; pad rows are zeroed once at staging so the
// WMMA inner loops are branch-free (no per-k EXEC save/restore, no clamps).
#define LDS_F_PER_WAVE (32*QK_STRIDE + 32*QK_STRIDE + 32*V_STRIDE + 32*P_STRIDE) // 7808 floats

__device__ __forceinline__ v8f wmma4(v2f a, v2f b, v8f c) {
  // V_WMMA_F32_16X16X4_F32: D = A(16x4) * B(4x16) + C(16x16)
  return __builtin_amdgcn_wmma_f32_16x16x4_f32(false, a, false, b, (short)0, c,
                                               false, false);
}

__global__ __launch_bounds__(WAVES_PER_WG * 32)
void deatt_wmma_kernel(const float* __restrict__ Q, const float* __restrict__ K,
                       const float* __restrict__ V, const float* __restrict__ M,
                       float* __restrict__ deAtt, float* __restrict__ attnOut,
                       int bh_total) {
  extern __shared__ float smem[];
  const int lane = threadIdx.x & 31;
  const int wave = threadIdx.x >> 5;
  const int bh   = blockIdx.x * WAVES_PER_WG + wave;
  if (bh >= bh_total) return;  // uniform per wave

  float* qs = smem + wave * LDS_F_PER_WAVE;
  float* ks = qs + 32 * QK_STRIDE;
  float* vs = ks + 32 * QK_STRIDE;
  float* ps = vs + 32 * V_STRIDE;

  const float* Qg = Q + (size_t)bh * (SEQ * DK);
  const float* Kg = K + (size_t)bh * (SEQ * DK);
  const float* Vg = V + (size_t)bh * (SEQ * DK);
  const float* Mg = M + (size_t)bh * (SEQ * SEQ);

  // ---- Zero pad rows 20..31 once (makes all WMMA fragment loads unconditional) ----
  for (int i = lane; i < 12 * QK_STRIDE; i += 32) {
    qs[20 * QK_STRIDE + i] = 0.f;
    ks[20 * QK_STRIDE + i] = 0.f;
  }
  for (int i = lane; i < 12 * V_STRIDE; i += 32) vs[20 * V_STRIDE + i] = 0.f;

  // ---- Stage Q,K,V -> LDS with coalesced float4 loads (10 iters each) ----
  for (int i = lane; i < (SEQ * DK) / 4; i += 32) {
    float4 q4 = reinterpret_cast<const float4*>(Qg)[i];
    float4 k4 = reinterpret_cast<const float4*>(Kg)[i];
    float4 v4 = reinterpret_cast<const float4*>(Vg)[i];
    int r = i >> 4;            // row = i*4/64
    int c = (i & 15) << 2;     // col
    float* qd = &qs[r * QK_STRIDE + c];
    float* kd = &ks[r * QK_STRIDE + c];
    float* vd = &vs[r * V_STRIDE + c];
    qd[0] = q4.x; qd[1] = q4.y; qd[2] = q4.z; qd[3] = q4.w;
    kd[0] = k4.x; kd[1] = k4.y; kd[2] = k4.z; kd[3] = k4.w;
    vd[0] = v4.x; vd[1] = v4.y; vd[2] = v4.z; vd[3] = v4.w;
  }
  // wave-private LDS slice: no barrier needed, DScnt waits are inserted by compiler

  const int half = lane >> 4;      // 0: K={d0,d0+1}, 1: K={d0+2,d0+3}
  const int lr   = lane & 15;

  // ---- GEMM1: S[32,32] = Qpad[32,64] * Kpad^T[64,32]  (64 WMMA, branch-free) ----
  const float* qrow0 = &qs[lr * QK_STRIDE];
  const float* qrow1 = &qs[(16 + lr) * QK_STRIDE];
  const float* krow0 = &ks[lr * QK_STRIDE];
  const float* krow1 = &ks[(16 + lr) * QK_STRIDE];
  v8f c00 = {}, c01 = {}, c10 = {}, c11 = {};
#pragma unroll
  for (int d0 = 0; d0 < DK; d0 += 4) {
    const int col = d0 + 2 * half;
    v2f a0 = *(const v2f*)&qrow0[col];
    v2f a1 = *(const v2f*)&qrow1[col];
    v2f b0 = *(const v2f*)&krow0[col];
    v2f b1 = *(const v2f*)&krow1[col];
    c00 = wmma4(a0, b0, c00);
    c01 = wmma4(a0, b1, c01);
    c10 = wmma4(a1, b0, c10);
    c11 = wmma4(a1, b1, c11);
  }

  // ---- scores = exp(S/8) * mask (zero outside 20x20) ----
  const float scale = 0.125f;
  {
    v8f* tiles[4] = {&c00, &c01, &c10, &c11};
    const int m0s[4] = {0, 0, 16, 16};
    const int n0s[4] = {0, 16, 0, 16};
#pragma unroll
    for (int t = 0; t < 4; ++t) {
      const int n = n0s[t] + lr;
#pragma unroll
      for (int j = 0; j < 8; ++j) {
        const int m = m0s[t] + j + 8 * half;
        float s = 0.f;
        if (m < SEQ && n < SEQ)
          s = __expf((*tiles[t])[j] * scale) * Mg[m * SEQ + n];
        (*tiles[t])[j] = s;
      }
    }
  }

  // ---- row-sum over N (each 16-lane half holds one row per accumulator VGPR) ----
#pragma unroll
  for (int j = 0; j < 8; ++j) {
    float t0 = c00[j] + c01[j];   // invalid n already zeroed
    float t1 = c10[j] + c11[j];
    t0 += __shfl_xor(t0, 1); t1 += __shfl_xor(t1, 1);
    t0 += __shfl_xor(t0, 2); t1 += __shfl_xor(t1, 2);
    t0 += __shfl_xor(t0, 4); t1 += __shfl_xor(t1, 4);
    t0 += __shfl_xor(t0, 8); t1 += __shfl_xor(t1, 8);
    const float i0 = 1.f / (t0 + 1e-8f);
    const float i1 = 1.f / (t1 + 1e-8f);
    c00[j] *= i0; c01[j] *= i0;
    c10[j] *= i1; c11[j] *= i1;
  }

  // ---- store attn; build P = attn + Md (Md[m][n] = -|m-n|), zero-padded 32x32 ----
  float* Ag = attnOut + (size_t)bh * (SEQ * SEQ);
  {
    v8f* tiles[4] = {&c00, &c01, &c10, &c11};
    const int m0s[4] = {0, 0, 16, 16};
    const int n0s[4] = {0, 16, 0, 16};
#pragma unroll
    for (int t = 0; t < 4; ++t) {
      const int n = n0s[t] + lr;
#pragma unroll
      for (int j = 0; j < 8; ++j) {
        const int m = m0s[t] + j + 8 * half;
        const float a = (*tiles[t])[j];
        float p = 0.f;
        if (m < SEQ && n < SEQ) {
          Ag[m * SEQ + n] = a;
          p = a - fabsf((float)(m - n));
        }
        ps[m * P_STRIDE + n] = p;  // full 32x32 written -> GEMM2 needs no selects
      }
    }
  }

  // ---- GEMM2: D[32,64] = P[32,32] * Vpad[32,64]  (64 WMMA, branch-free) ----
  v8f acc[2][4];
#pragma unroll
  for (int mt = 0; mt < 2; ++mt)
#pragma unroll
    for (int nt = 0; nt < 4; ++nt) acc[mt][nt] = {};

  const float* prow0 = &ps[lr * P_STRIDE];
  const float* prow1 = &ps[(16 + lr) * P_STRIDE];
#pragma unroll
  for (int k0 = 0; k0 < 32; k0 += 4) {
    const int kr = k0 + 2 * half;
    v2f a0 = *(const v2f*)&prow0[kr];
    v2f a1 = *(const v2f*)&prow1[kr];
#pragma unroll
    for (int nt = 0; nt < 4; ++nt) {
      const int d = nt * 16 + lr;
      v2f b = { vs[kr * V_STRIDE + d], vs[(kr + 1) * V_STRIDE + d] };
      acc[0][nt] = wmma4(a0, b, acc[0][nt]);
      acc[1][nt] = wmma4(a1, b, acc[1][nt]);
    }
  }

  // ---- store deAtt rows m < 20 ----
  float* Og = deAtt + (size_t)bh * (SEQ * DK);
#pragma unroll
  for (int mt = 0; mt < 2; ++mt)
#pragma unroll
    for (int nt = 0; nt < 4; ++nt) {
      const int d = nt * 16 + lr;
#pragma unroll
      for (int j = 0; j < 8; ++j) {
        const int m = mt * 16 + j + 8 * half;
        if (m < SEQ) Og[m * DK + d] = acc[mt][nt][j];
      }
    }
}

extern "C" void kernel_launch(void* const* d_in, const int* in_sizes, int n_in,
                              void* d_out, int out_size, void* d_ws, size_t ws_size,
                              hipStream_t stream) {
  const float* Q = (const float*)d_in[0];
  const float* K = (const float*)d_in[1];
  const float* V = (const float*)d_in[2];
  const float* M = (const float*)d_in[3];

  const int bh_total = in_sizes[0] / (SEQ * DK);  // 8192 for B=512,H=16
  float* deAtt = (float*)d_out;
  float* attn  = deAtt + (size_t)bh_total * SEQ * DK;

  dim3 grid((bh_total + WAVES_PER_WG - 1) / WAVES_PER_WG);
  dim3 block(WAVES_PER_WG * 32);
  const size_t shmem = (size_t)WAVES_PER_WG * LDS_F_PER_WAVE * sizeof(float);
  hipLaunchKernelGGL(deatt_wmma_kernel, grid, block, shmem, stream,
                     Q, K, V, M, deAtt, attn, bh_total);
}